// CADefnAttn_76768245449529
// MI455X (gfx1250) — compile-verified
//
#include <hip/hip_runtime.h>
#include <hip/hip_bf16.h>
#include <math.h>

// ---------------------------------------------------------------------------
// BEVFormer-style spatial cross (deformable) attention for MI455X / gfx1250.
// All dense projections use V_WMMA_F32_16X16X4_F32 (fp32 in/out, wave32).
// ---------------------------------------------------------------------------

typedef __attribute__((ext_vector_type(2))) float v2f;
typedef __attribute__((ext_vector_type(8))) float v8f;

// Problem constants (from reference setup_inputs)
#define BB    2
#define NCAM  6
#define NQ    10000
#define CC    128
#define YY    4
#define HH    4      // heads
#define PP    8      // points
#define DD    32     // head dim
#define HS    32
#define WS    88
#define HW    (HS * WS)        // 2816

// Workspace layout (floats)
#define VALP_OFF  0ull
#define VALP_ELEMS ((size_t)BB * NCAM * HW * CC)                 // 4,325,376
#define OFF_OFF   (VALP_OFF + VALP_ELEMS)
#define OFF_ELEMS ((size_t)BB * NQ * HH * PP * 2)                // 1,280,000
#define AW_OFF    (OFF_OFF + OFF_ELEMS)
#define AW_ELEMS  ((size_t)BB * NQ * HH * PP)                    //   640,000
#define ACC_OFF   (AW_OFF + AW_ELEMS)
#define ACC_ELEMS ((size_t)BB * NQ * CC)                         // 2,560,000

// ---------------------------------------------------------------------------
// Kernel 1: value projection.  valp[bn][s][co] = value[n][s][b][:] . W_val[co][:] + b_val[co]
// One wave (32 lanes) computes one 16x16 output tile via 32x V_WMMA_F32_16X16X4_F32.
// grid = (C/16, HW/16, 12), block = 32
// ---------------------------------------------------------------------------
__global__ __launch_bounds__(32)
void k_val_proj(const float* __restrict__ value, const float* __restrict__ W_val,
                const float* __restrict__ b_val, float* __restrict__ valp) {
    const int lane  = threadIdx.x;
    const int lrow  = lane & 15;     // M (for A) / N (for B,D)
    const int khalf = lane >> 4;     // 0 -> K{0,1}, 1 -> K{2,3}
    const int col0  = blockIdx.x * 16;
    const int row0  = blockIdx.y * 16;
    const int bn    = blockIdx.z;
    const int bb    = bn / NCAM, nn = bn % NCAM;

    // value strides: [n][s][b][c]  => ((n*HW + s)*BB + b)*CC + c
    const float* arow = value + (((size_t)nn * HW + (row0 + lrow)) * BB + bb) * CC;
    const float* brow = W_val + (size_t)(col0 + lrow) * CC;   // B[k][n] = W_val[col][k]

    v8f acc = {};
    for (int k0 = 0; k0 < CC; k0 += 4) {
        const int ka = k0 + khalf * 2;
        v2f a; a.x = arow[ka]; a.y = arow[ka + 1];
        v2f b; b.x = brow[ka]; b.y = brow[ka + 1];
        acc = __builtin_amdgcn_wmma_f32_16x16x4_f32(false, a, false, b,
                                                    (short)0, acc, false, false);
    }
    const int col = col0 + lrow;
    const float bias = b_val[col];
    for (int r = 0; r < 8; ++r) {
        const int row = row0 + r + khalf * 8;
        valp[((size_t)bn * HW + row) * CC + col] = acc[r] + bias;
    }
}

// ---------------------------------------------------------------------------
// Kernel 2: fused offset+attention projection (per batch, NOT per camera —
// the reference broadcasts q over cameras, so off/aw depend only on b).
// Output cols 0..63 -> off (W_off,b_off), 64..95 -> raw attn (W_attn,b_attn).
// grid = (6, NQ/16, 2), block = 32
// ---------------------------------------------------------------------------
__global__ __launch_bounds__(32)
void k_offattn_proj(const float* __restrict__ query, const float* __restrict__ query_pos,
                    const float* __restrict__ W_off, const float* __restrict__ b_off,
                    const float* __restrict__ W_attn, const float* __restrict__ b_attn,
                    float* __restrict__ off_out, float* __restrict__ aw_out) {
    const int lane  = threadIdx.x;
    const int lrow  = lane & 15;
    const int khalf = lane >> 4;
    const int col0  = blockIdx.x * 16;
    const int row0  = blockIdx.y * 16;
    const int bb    = blockIdx.z;

    const size_t qoff = ((size_t)bb * NQ + (row0 + lrow)) * CC;
    const int col = col0 + lrow;
    const float* brow = (col < 64) ? (W_off + (size_t)col * CC)
                                   : (W_attn + (size_t)(col - 64) * CC);

    v8f acc = {};
    for (int k0 = 0; k0 < CC; k0 += 4) {
        const int ka = k0 + khalf * 2;
        v2f a;
        a.x = query[qoff + ka]     + query_pos[qoff + ka];
        a.y = query[qoff + ka + 1] + query_pos[qoff + ka + 1];
        v2f b; b.x = brow[ka]; b.y = brow[ka + 1];
        acc = __builtin_amdgcn_wmma_f32_16x16x4_f32(false, a, false, b,
                                                    (short)0, acc, false, false);
    }
    const float bias = (col < 64) ? b_off[col] : b_attn[col - 64];
    for (int r = 0; r < 8; ++r) {
        const int row = row0 + r + khalf * 8;
        const float v = acc[r] + bias;
        if (col < 64)
            off_out[((size_t)bb * NQ + row) * 64 + col] = v;
        else
            aw_out[((size_t)bb * NQ + row) * 32 + (col - 64)] = v;
    }
}

// ---------------------------------------------------------------------------
// Kernel 3: per-(b,q,head) softmax over L*P = 8 points, in place on aw.
// ---------------------------------------------------------------------------
__global__ __launch_bounds__(256)
void k_softmax(float* __restrict__ aw) {
    const int idx = blockIdx.x * blockDim.x + threadIdx.x;
    if (idx >= BB * NQ * HH) return;
    const int h  = idx & (HH - 1);
    const int q  = (idx / HH) % NQ;
    const int bb = idx / (HH * NQ);
    float* p = aw + ((size_t)bb * NQ + q) * (HH * PP) + h * PP;
    float mx = p[0];
    for (int i = 1; i < PP; ++i) mx = fmaxf(mx, p[i]);
    float e[PP], s = 0.f;
    for (int i = 0; i < PP; ++i) { e[i] = expf(p[i] - mx); s += e[i]; }
    const float inv = 1.f / s;
    for (int i = 0; i < PP; ++i) p[i] = e[i] * inv;
}

// ---------------------------------------------------------------------------
// Kernel 4: deformable bilinear sampling + masked camera mean.
// One 128-thread block per (query, batch); lane = output channel (h*32+d),
// so each corner gather is a coalesced 128B segment per head. valp (17 MB)
// lives entirely in the 192 MB L2, so gathers hit L2.
// acc[bb][q][c] = (1/max(cnt,1)) * sum_{masked n} sum_{p, h=c/32} aw * bilinear(valp)
// ---------------------------------------------------------------------------
__global__ __launch_bounds__(128)
void k_deform(const float* __restrict__ valp, const float* __restrict__ off,
              const float* __restrict__ aw, const float* __restrict__ ref_pts_cam,
              const unsigned char* __restrict__ bev_mask, float* __restrict__ acc) {
    const int q  = blockIdx.x;
    const int bb = blockIdx.y;
    const int h  = threadIdx.x >> 5;   // head
    const int d  = threadIdx.x & 31;   // channel within head

    const float* offq = off + ((size_t)bb * NQ + q) * 64 + h * (PP * 2);
    const float* awq  = aw  + ((size_t)bb * NQ + q) * 32 + h * PP;

    float accv = 0.f;
    int cnt = 0;
    for (int nn = 0; nn < NCAM; ++nn) {
        // m = any over y of bev_mask[nn][bb][q][0..3]; bool bytes -> u32 test
        const unsigned int mw = *(const unsigned int*)(bev_mask +
                                 (((size_t)nn * BB + bb) * NQ + q) * YY);
        if (mw == 0u) continue;
        ++cnt;
        const int bn = bb * NCAM + nn;
        const float* vbase = valp + (size_t)bn * HW * CC + h * DD + d;
        const float* rp = ref_pts_cam + (((size_t)nn * BB + bb) * NQ + q) * (YY * 2);
        float camacc = 0.f;
        for (int p = 0; p < PP; ++p) {
            const int yi = p & 3;                       // p = pi*4 + yi (reshape order)
            const float lx = rp[yi * 2 + 0] + offq[p * 2 + 0] * (1.f / (float)WS);
            const float ly = rp[yi * 2 + 1] + offq[p * 2 + 1] * (1.f / (float)HS);
            const float x  = lx * (float)WS - 0.5f;
            const float yv = ly * (float)HS - 0.5f;
            const float x0f = floorf(x), y0f = floorf(yv);
            const int   x0 = (int)x0f,  y0 = (int)y0f;
            const float fx = x - x0f,   fy = yv - y0f;
            float s = 0.f;
            #pragma unroll
            for (int dy = 0; dy < 2; ++dy) {
                const int yc = y0 + dy;
                if (yc < 0 || yc >= HS) continue;
                const float wy = dy ? fy : (1.f - fy);
                #pragma unroll
                for (int dx = 0; dx < 2; ++dx) {
                    const int xc = x0 + dx;
                    if (xc < 0 || xc >= WS) continue;
                    const float wgt = wy * (dx ? fx : (1.f - fx));
                    s += wgt * vbase[((size_t)yc * WS + xc) * CC];
                }
            }
            camacc += awq[p] * s;
        }
        accv += camacc;
    }
    const float inv = 1.f / (float)(cnt > 0 ? cnt : 1);
    acc[((size_t)bb * NQ + q) * CC + h * DD + d] = accv * inv;
}

// ---------------------------------------------------------------------------
// Kernel 5: output projection + bias + residual.
// out[row][col] = acc[row][:] . W_out[col][:] + b_out[col] + query[row][col]
// rows flat over (b, Nq). grid = (C/16, 20000/16), block = 32
// ---------------------------------------------------------------------------
__global__ __launch_bounds__(32)
void k_out_proj(const float* __restrict__ acc_in, const float* __restrict__ W_out,
                const float* __restrict__ b_out, const float* __restrict__ query,
                float* __restrict__ out) {
    const int lane  = threadIdx.x;
    const int lrow  = lane & 15;
    const int khalf = lane >> 4;
    const int col0  = blockIdx.x * 16;
    const int row0  = blockIdx.y * 16;

    const float* arow = acc_in + (size_t)(row0 + lrow) * CC;
    const float* brow = W_out + (size_t)(col0 + lrow) * CC;

    v8f acc = {};
    for (int k0 = 0; k0 < CC; k0 += 4) {
        const int ka = k0 + khalf * 2;
        v2f a; a.x = arow[ka]; a.y = arow[ka + 1];
        v2f b; b.x = brow[ka]; b.y = brow[ka + 1];
        acc = __builtin_amdgcn_wmma_f32_16x16x4_f32(false, a, false, b,
                                                    (short)0, acc, false, false);
    }
    const int col = col0 + lrow;
    const float bias = b_out[col];
    for (int r = 0; r < 8; ++r) {
        const int row = row0 + r + khalf * 8;
        out[(size_t)row * CC + col] = acc[r] + bias + query[(size_t)row * CC + col];
    }
}

// ---------------------------------------------------------------------------
extern "C" void kernel_launch(void* const* d_in, const int* in_sizes, int n_in,
                              void* d_out, int out_size, void* d_ws, size_t ws_size,
                              hipStream_t stream) {
    (void)in_sizes; (void)n_in; (void)out_size; (void)ws_size;

    const float* query       = (const float*)d_in[0];
    // d_in[1] = key (unused by the reference math)
    const float* value       = (const float*)d_in[2];
    const float* query_pos   = (const float*)d_in[3];
    const float* ref_pts_cam = (const float*)d_in[4];
    // d_in[5] = spatial_shapes (constant [[32, 88]], hardcoded)
    const unsigned char* bev_mask = (const unsigned char*)d_in[6];
    const float* W_off  = (const float*)d_in[7];
    const float* b_off  = (const float*)d_in[8];
    const float* W_attn = (const float*)d_in[9];
    const float* b_attn = (const float*)d_in[10];
    const float* W_val  = (const float*)d_in[11];
    const float* b_val  = (const float*)d_in[12];
    const float* W_out  = (const float*)d_in[13];
    const float* b_out  = (const float*)d_in[14];

    float* ws   = (float*)d_ws;
    float* valp = ws + VALP_OFF;
    float* off  = ws + OFF_OFF;
    float* aw   = ws + AW_OFF;
    float* acc  = ws + ACC_OFF;
    float* out  = (float*)d_out;

    // 1) value projection: 12 x (2816 x 128) tiles of 16x16
    k_val_proj<<<dim3(CC / 16, HW / 16, BB * NCAM), 32, 0, stream>>>(value, W_val, b_val, valp);

    // 2) offset + attention projection (per batch only; q broadcast over cams)
    k_offattn_proj<<<dim3(96 / 16, NQ / 16, BB), 32, 0, stream>>>(
        query, query_pos, W_off, b_off, W_attn, b_attn, off, aw);

    // 3) per-head softmax over 8 points
    k_softmax<<<(BB * NQ * HH + 255) / 256, 256, 0, stream>>>(aw);

    // 4) deformable sampling + masked camera mean
    k_deform<<<dim3(NQ, BB), 128, 0, stream>>>(valp, off, aw, ref_pts_cam, bev_mask, acc);

    // 5) output projection + bias + residual
    k_out_proj<<<dim3(CC / 16, (BB * NQ) / 16), 32, 0, stream>>>(acc, W_out, b_out, query, out);
}